// KMeans_39350490366326
// MI455X (gfx1250) — compile-verified
//
#include <hip/hip_runtime.h>

// Problem constants (match reference)
#define B_  8192
#define P_  64
#define T_  406
#define D_  10
#define TD_ 4060   // T*D, K-dim of the cross GEMM (= 20 * 203)
#define TP_ 408    // T padded to multiple of 4 for the cp2 GEMM
#define NCHUNK_ 203  // TD_ / 20

typedef float v2f __attribute__((ext_vector_type(2)));
typedef float v8f __attribute__((ext_vector_type(8)));

// ---------------------------------------------------------------------------
// Kernel 1: cp2[p, t] = sum_d proto[p,t,d]^2, laid out [P][TP_] zero-padded.
// ---------------------------------------------------------------------------
__global__ __launch_bounds__(256) void cp2_kernel(const float* __restrict__ proto,
                                                  float* __restrict__ cp2) {
  int i = blockIdx.x * 256 + threadIdx.x;
  if (i >= P_ * TP_) return;
  int p = i / TP_;
  int t = i - p * TP_;
  float s = 0.f;
  if (t < T_) {
    const float* pr = proto + (size_t)p * TD_ + (size_t)t * D_;
#pragma unroll
    for (int d = 0; d < D_; ++d) s += pr[d] * pr[d];
  }
  cp2[i] = s;
}

// ---------------------------------------------------------------------------
// Kernel 2: per workgroup (1 wave32) compute a 16(batch) x 64(protos) tile of
//   dist[b,p] = sum m*x^2 - 2*sum m*x*c + sum m*c^2
// via V_WMMA_F32_16X16X4_F32, then argmin over p.
//
// A fragment (16x4 f32): lane = 16*kp + m holds K = {2kp, 2kp+1} of row m.
// B fragment (4x16 f32): lane = 16*kp + n holds K = {2kp, 2kp+1} of col n.
// C/D (16x16 f32): VGPR j, lanes 0-15 -> M=j, lanes 16-31 -> M=j+8; N=lane&15.
//
// Phase-1 K loop runs in chunks of 20 (lcm of K-step 4 and D=10) so each lane
// needs only mask[t0] / mask[t0+1] per chunk; the t-selection is compile-time
// per sub-step except j==2 (one v_cndmask on the lane's K-half).
// ---------------------------------------------------------------------------
__global__ __launch_bounds__(32) void dist_kernel(
    const float* __restrict__ x, const float* __restrict__ mask,
    const float* __restrict__ proto, const float* __restrict__ cp2,
    float* __restrict__ dist_out, float* __restrict__ idxf,
    int* __restrict__ idxi)
{
  __shared__ float sqx[16];
  __shared__ float dtile[16 * P_];

  const int lane = threadIdx.x;   // 0..31
  const int m    = lane & 15;     // row (A) / col (B) within 16-tile
  const int kp   = lane >> 4;     // K sub-pair: 0 -> K={0,1}, 1 -> K={2,3}
  const int b0   = blockIdx.x * 16;

  const float* xrow = x    + (size_t)(b0 + m) * TD_;
  const float* mrow = mask + (size_t)(b0 + m) * T_;
  const float* p0   = proto + (size_t)( 0 + m) * TD_;
  const float* p1   = proto + (size_t)(16 + m) * TD_;
  const float* p2   = proto + (size_t)(32 + m) * TD_;
  const float* p3   = proto + (size_t)(48 + m) * TD_;

  v8f acc[4] = {v8f{}, v8f{}, v8f{}, v8f{}};
  float s = 0.f;   // per-lane partial of sum m*x^2

  // Phase 1: acc += (-2 * m * x) * c^T   over K = 4060, chunked by 20.
  for (int c = 0; c < NCHUNK_; ++c) {
    // Two mask values cover this whole 20-wide K-chunk for every lane.
    const v2f mpair = *(const v2f*)(mrow + 2 * c);   // 8B aligned (2c even)
    const float mA = mpair.x;
    const float mB = mpair.y;
    const float mSel = kp ? mB : mA;                 // only j==2 needs this
    const int kbase = 20 * c + 2 * kp;

#pragma unroll
    for (int j = 0; j < 5; ++j) {
      const int k = kbase + 4 * j;
      // Element t-offsets (4j+2kp)/10 and (4j+2kp+1)/10 are equal and
      // compile-time resolvable except at j==2 (handled by mSel).
      const float mm = (j < 2) ? mA : ((j == 2) ? mSel : mB);

      v2f xv = *(const v2f*)(xrow + k);
      s += mm * (xv.x * xv.x + xv.y * xv.y);
      v2f a; a.x = -2.f * mm * xv.x; a.y = -2.f * mm * xv.y;

      v2f bv0 = *(const v2f*)(p0 + k);
      v2f bv1 = *(const v2f*)(p1 + k);
      v2f bv2 = *(const v2f*)(p2 + k);
      v2f bv3 = *(const v2f*)(p3 + k);
      acc[0] = __builtin_amdgcn_wmma_f32_16x16x4_f32(false, a, false, bv0, (short)0, acc[0], false, false);
      acc[1] = __builtin_amdgcn_wmma_f32_16x16x4_f32(false, a, false, bv1, (short)0, acc[1], false, false);
      acc[2] = __builtin_amdgcn_wmma_f32_16x16x4_f32(false, a, false, bv2, (short)0, acc[2], false, false);
      acc[3] = __builtin_amdgcn_wmma_f32_16x16x4_f32(false, a, false, bv3, (short)0, acc[3], false, false);
    }
  }

  // Phase 2: acc += mask * cp2^T   over K = 408 (cp2 zero-padded)
  const float* c0 = cp2 + (size_t)( 0 + m) * TP_;
  const float* c1 = cp2 + (size_t)(16 + m) * TP_;
  const float* c2 = cp2 + (size_t)(32 + m) * TP_;
  const float* c3 = cp2 + (size_t)(48 + m) * TP_;
  for (int k0 = 0; k0 < TP_; k0 += 4) {
    const int k = k0 + 2 * kp;
    v2f a;
    a.x = (k     < T_) ? mrow[k]     : 0.f;
    a.y = (k + 1 < T_) ? mrow[k + 1] : 0.f;
    v2f bv0 = *(const v2f*)(c0 + k);
    v2f bv1 = *(const v2f*)(c1 + k);
    v2f bv2 = *(const v2f*)(c2 + k);
    v2f bv3 = *(const v2f*)(c3 + k);
    acc[0] = __builtin_amdgcn_wmma_f32_16x16x4_f32(false, a, false, bv0, (short)0, acc[0], false, false);
    acc[1] = __builtin_amdgcn_wmma_f32_16x16x4_f32(false, a, false, bv1, (short)0, acc[1], false, false);
    acc[2] = __builtin_amdgcn_wmma_f32_16x16x4_f32(false, a, false, bv2, (short)0, acc[2], false, false);
    acc[3] = __builtin_amdgcn_wmma_f32_16x16x4_f32(false, a, false, bv3, (short)0, acc[3], false, false);
  }

  // Reduce sq_x across the two K-halves (lanes m and m+16), publish via LDS.
  s += __shfl_xor(s, 16, 32);
  if (lane < 16) sqx[lane] = s;
  __syncthreads();

  // Emit distances (global + LDS tile for argmin).
#pragma unroll
  for (int nt = 0; nt < 4; ++nt) {
#pragma unroll
    for (int j = 0; j < 8; ++j) {
      const int M = j + 8 * kp;        // row in 16-tile
      const int N = m + 16 * nt;       // prototype index
      float v = acc[nt][j] + sqx[M];
      dtile[M * P_ + N] = v;
      dist_out[(size_t)(b0 + M) * P_ + N] = v;
    }
  }
  __syncthreads();

  // Argmin per row (first-min tie-break like jnp.argmin).
  if (lane < 16) {
    const float* row = dtile + lane * P_;
    float best = row[0];
    int bi = 0;
    for (int p = 1; p < P_; ++p) {
      float v = row[p];
      if (v < best) { best = v; bi = p; }
    }
    idxf[b0 + lane] = (float)bi;
    idxi[b0 + lane] = bi;
  }
}

// ---------------------------------------------------------------------------
// Kernel 3: output_seq = proto[idx[b]] gather (prototypes live in L2) and
//           input_seq pass-through copy. Pure streaming.
// ---------------------------------------------------------------------------
__global__ __launch_bounds__(256) void gather_kernel(
    const float* __restrict__ x, const float* __restrict__ proto,
    const int* __restrict__ idx, float* __restrict__ out_seq,
    float* __restrict__ out_in, int n)
{
  int i = blockIdx.x * 256 + threadIdx.x;
  if (i >= n) return;
  int b = i / TD_;
  int r = i - b * TD_;
  out_seq[i] = proto[(size_t)idx[b] * TD_ + r];
  out_in[i]  = x[i];
}

// ---------------------------------------------------------------------------
// Kernel 4: mask + label pass-throughs (label int -> float).
// ---------------------------------------------------------------------------
__global__ __launch_bounds__(256) void tail_kernel(
    const float* __restrict__ mask, const int* __restrict__ label,
    float* __restrict__ out_mask, float* __restrict__ out_label,
    int nm, int nb)
{
  int i = blockIdx.x * 256 + threadIdx.x;
  if (i < nm) out_mask[i] = mask[i];
  if (i < nb) out_label[i] = (float)label[i];
}

// ---------------------------------------------------------------------------
extern "C" void kernel_launch(void* const* d_in, const int* in_sizes, int n_in,
                              void* d_out, int out_size, void* d_ws, size_t ws_size,
                              hipStream_t stream) {
  (void)in_sizes; (void)n_in; (void)out_size; (void)ws_size;

  const float* x     = (const float*)d_in[0];
  const int*   label = (const int*)d_in[1];
  const float* mask  = (const float*)d_in[2];
  const float* proto = (const float*)d_in[3];
  float* out = (float*)d_out;

  // Output layout (floats), concatenated in reference return order.
  const size_t o_seq  = 0;
  const size_t o_in   = o_seq  + (size_t)B_ * TD_;
  const size_t o_dist = o_in   + (size_t)B_ * TD_;
  const size_t o_idx  = o_dist + (size_t)B_ * P_;
  const size_t o_lab  = o_idx  + (size_t)B_;
  const size_t o_msk  = o_lab  + (size_t)B_;

  // Workspace: cp2 table [P][TP_] + int indices [B].
  float* cp2  = (float*)d_ws;
  int*   idxi = (int*)((char*)d_ws + (size_t)P_ * TP_ * sizeof(float));

  cp2_kernel<<<(P_ * TP_ + 255) / 256, 256, 0, stream>>>(proto, cp2);

  dist_kernel<<<B_ / 16, 32, 0, stream>>>(x, mask, proto, cp2,
                                          out + o_dist, out + o_idx, idxi);

  const int nbig = B_ * TD_;
  gather_kernel<<<(nbig + 255) / 256, 256, 0, stream>>>(x, proto, idxi,
                                                        out + o_seq, out + o_in, nbig);

  tail_kernel<<<(B_ * T_ + 255) / 256, 256, 0, stream>>>(mask, label,
                                                         out + o_msk, out + o_lab,
                                                         B_ * T_, B_);
}